// LSTM_CRF_50921132262039
// MI455X (gfx1250) — compile-verified
//
#include <hip/hip_runtime.h>
#include <hip/hip_bf16.h>
#include <math.h>

// ---------------------------------------------------------------------------
// LSTM-CRF for MI455X (gfx1250, wave32).
//   Stage 0: zero a scratch "zero row" used to de-predicate padded N-tiles.
//   Stage 1: embed gather via TDM gather-mode (tensor_load_to_lds, 16 row
//            indices) + input-gate GEMM (f32 WMMA 16x16x4), biases folded.
//   Stage 2: bidirectional LSTM recurrence, batch tiled 16-wide, Whh/h/c in
//            LDS (Whh zero-padded to 208 rows -> no predication in k-loop),
//            per-step gates via f32 WMMA, global_prefetch of next XG tile.
//   Stage 3: FC -> logits via f32 WMMA (N padded to 16, zero-row pointer).
//   Stage 4: CRF forward algorithm, one wave per batch row, shfl logsumexp.
//   Stage 5: reduce ll[128] -> loss (d_out[0]).
// ---------------------------------------------------------------------------

typedef __attribute__((ext_vector_type(2))) float v2f;
typedef __attribute__((ext_vector_type(8))) float v8f;
typedef __attribute__((ext_vector_type(4))) unsigned int v4u;
typedef __attribute__((ext_vector_type(4))) int v4i;
typedef __attribute__((ext_vector_type(8))) int v8i;

#define Bsz   128
#define Tlen  256
#define Edim  300
#define Hdim  50
#define G4H   200          // 4*H
#define Cdim  9
#define Vsz   30000
#define Mrows (Bsz*Tlen)   // 32768

#if __has_builtin(__builtin_amdgcn_tensor_load_to_lds) && (__clang_major__ < 23)
#define GATES_TDM 1
#endif

// ---------------------------------------------------------------------------
// Kernel 0: zero the shared zero-row (used as B source for padded lanes).
// ---------------------------------------------------------------------------
__global__ void __launch_bounds__(512)
zrow_kernel(float* __restrict__ z) { z[threadIdx.x] = 0.f; }

// ---------------------------------------------------------------------------
// Kernel 1: XG[m, g] = embed[ids[m]] . Wih[g, :]  + (bih[g] + bhh[g])
// One block = one 16-row M tile; TDM gather stages the 16 embedding rows
// (16x300 f32, dense) into LDS in a single tensor op.
// ---------------------------------------------------------------------------
__global__ void __launch_bounds__(128)
gates_kernel(const int* __restrict__ ids, const float* __restrict__ emb,
             const float* __restrict__ WihF, const float* __restrict__ bihF,
             const float* __restrict__ bhhF,
             const float* __restrict__ WihB, const float* __restrict__ bihB,
             const float* __restrict__ bhhB,
             const float* __restrict__ zrow,
             float* __restrict__ XGF, float* __restrict__ XGB) {
  __shared__ __align__(16) float lA[16][Edim];   // dense; pattern is bank-clean

  const int tid  = threadIdx.x;
  const int wave = tid >> 5;
  const int lane = tid & 31;
  const int half = lane >> 4;            // 0: K,K+1   1: K+2,K+3
  const int ln   = lane & 15;
  const size_t mbase = (size_t)blockIdx.x * 16;

#ifdef GATES_TDM
  // ---- Tensor Data Mover gather: 16 rows of embed_table -> lA ----
  unsigned pk[8];
#pragma unroll
  for (int i = 0; i < 8; ++i) {
    unsigned lo = (unsigned)ids[mbase + 2 * i];
    unsigned hi = (unsigned)ids[mbase + 2 * i + 1];
    pk[i] = (lo & 0xFFFFu) | (hi << 16);           // 16-bit row indices
  }
  {
    unsigned lds_base = (unsigned)(uintptr_t)(&lA[0][0]);
    unsigned long long gbase = (unsigned long long)(uintptr_t)emb;
    v4u g0;
    g0.x = 0x80000001u;                              // count=1 | gather_mode
    g0.y = lds_base;                                 // lds_addr (bytes)
    g0.z = (unsigned)(gbase & 0xFFFFFFFFu);          // global_addr[31:0]
    g0.w = (unsigned)((gbase >> 32) & 0x01FFFFFFu)   // global_addr[56:32]
         | (2u << 30);                               // type = 2 (image)
    v8i g1;
    g1[0] = (int)(2u << 16);          // data_size = 4B
    g1[1] = (int)((unsigned)Edim << 16); // tensor_dim0[15:0] @ bit48
    g1[2] = (int)((unsigned)Vsz << 16);  // tensor_dim1[15:0] @ bit80
    g1[3] = (int)((unsigned)Edim << 16); // tile_dim0 @ bit112
    g1[4] = 16;                        // tile_dim1 = #valid gather indices
    g1[5] = Edim;                      // tensor_dim0_stride[31:0]
    g1[6] = 0;
    g1[7] = 0;
    v4i g2, g3;
    g2.x = (int)pk[0]; g2.y = (int)pk[1]; g2.z = (int)pk[2]; g2.w = (int)pk[3];
    g3.x = (int)pk[4]; g3.y = (int)pk[5]; g3.z = (int)pk[6]; g3.w = (int)pk[7];
    if (wave == 0) {
      __builtin_amdgcn_tensor_load_to_lds(g0, g1, g2, g3, 0);
      __builtin_amdgcn_s_wait_tensorcnt(0);
    }
  }
  __syncthreads();
#else
  for (int idx = tid; idx < 16 * Edim; idx += 128) {
    int r = idx / Edim, e = idx % Edim;
    lA[r][e] = emb[(size_t)ids[mbase + r] * Edim + e];
  }
  __syncthreads();
#endif

  for (int nt = wave; nt < 26; nt += 4) {          // wave-uniform (EXEC full)
    const int dir  = nt / 13;
    const int gcol = (nt % 13) * 16 + ln;
    const bool gv  = gcol < G4H;
    const float* __restrict__ Wp = dir ? WihB : WihF;
    const float* __restrict__ Wrow = gv ? (Wp + (size_t)gcol * Edim) : zrow;

    v8f acc = {};
    for (int k = 0; k < Edim; k += 4) {            // 75 iters, no predication
      v2f a, b;
      a.x = lA[ln][k + 2 * half];
      a.y = lA[ln][k + 2 * half + 1];
      b.x = Wrow[k + 2 * half];
      b.y = Wrow[k + 2 * half + 1];
      acc = __builtin_amdgcn_wmma_f32_16x16x4_f32(false, a, false, b,
                                                  (short)0, acc, false, false);
    }
    if (gv) {
      float bias = dir ? (bihB[gcol] + bhhB[gcol]) : (bihF[gcol] + bhhF[gcol]);
      float* __restrict__ XG = dir ? XGB : XGF;
#pragma unroll
      for (int r = 0; r < 8; ++r)
        XG[(mbase + r + 8 * half) * G4H + gcol] = acc[r] + bias;
    }
  }
}

// ---------------------------------------------------------------------------
// Kernel 2: LSTM recurrence. grid = (8 batch-tiles, 2 directions), 4 waves.
// Per step:  G = XG_t + H(16x50) @ Whh^T(50x200)  via WMMA, then gate math.
// sW zero-padded to 208 rows / 53 cols -> inner loop has zero predication.
// ---------------------------------------------------------------------------
__global__ void __launch_bounds__(128)
lstm_kernel(const float* __restrict__ XGF, const float* __restrict__ XGB,
            const float* __restrict__ WhhF, const float* __restrict__ WhhB,
            float* __restrict__ Hout) {
  __shared__ float sW[208][53];   // Whh, rows >=200 and cols >=50 are 0
  __shared__ float sH[16][53];    // hidden state, K-padded with zeros
  __shared__ float sC[16][53];    // cell state
  __shared__ float sG[16][209];   // gate buffer

  const int tid  = threadIdx.x;
  const int wave = tid >> 5;
  const int lane = tid & 31;
  const int half = lane >> 4;
  const int ln   = lane & 15;
  const int btile = blockIdx.x;   // 0..7
  const int dir   = blockIdx.y;   // 0 fwd / 1 bwd
  const float* __restrict__ W  = dir ? WhhB : WhhF;   // (200,50)
  const float* __restrict__ XG = dir ? XGB  : XGF;

  for (int idx = tid; idx < 208 * 53; idx += 128) {
    int r = idx / 53, c = idx % 53;
    sW[r][c] = (r < G4H && c < Hdim) ? W[r * Hdim + c] : 0.f;
  }
  for (int idx = tid; idx < 16 * 53; idx += 128) {
    int r = idx / 53, c = idx % 53;
    sH[r][c] = 0.f;
    sC[r][c] = 0.f;
  }
  __syncthreads();

  for (int step = 0; step < Tlen; ++step) {
    const int pos = dir ? (Tlen - 1 - step) : step;  // input/output position

    for (int nt = wave; nt < 13; nt += 4) {          // wave-uniform
      const int gcol = nt * 16 + ln;                 // <= 207, sW padded
      const bool gv  = gcol < G4H;

      if (step + 1 < Tlen) {                         // hide next step's XG
        int npos = dir ? (Tlen - 2 - step) : (step + 1);
        __builtin_prefetch(
            &XG[((size_t)(btile * 16 + ln) * Tlen + npos) * G4H + gcol], 0, 3);
      }

      v8f acc;
#pragma unroll
      for (int r = 0; r < 8; ++r) {
        int b = btile * 16 + r + 8 * half;
        acc[r] = gv ? XG[((size_t)b * Tlen + pos) * G4H + gcol] : 0.f;
      }
      for (int k = 0; k < 52; k += 4) {              // 13 iters, zero-padded K
        v2f a, b;
        a.x = sH[ln][k + 2 * half];
        a.y = sH[ln][k + 2 * half + 1];
        b.x = sW[gcol][k + 2 * half];
        b.y = sW[gcol][k + 2 * half + 1];
        acc = __builtin_amdgcn_wmma_f32_16x16x4_f32(false, a, false, b,
                                                    (short)0, acc, false, false);
      }
      if (gv) {
#pragma unroll
        for (int r = 0; r < 8; ++r) sG[r + 8 * half][gcol] = acc[r];
      }
    }
    __syncthreads();

    for (int idx = tid; idx < 16 * Hdim; idx += 128) {
      int r = idx / Hdim, j = idx % Hdim;
      float gi = sG[r][j];
      float gf = sG[r][Hdim + j];
      float gg = sG[r][2 * Hdim + j];
      float go = sG[r][3 * Hdim + j];
      float i_ = 1.f / (1.f + __expf(-gi));
      float f_ = 1.f / (1.f + __expf(-gf));
      float o_ = 1.f / (1.f + __expf(-go));
      float cn = f_ * sC[r][j] + i_ * tanhf(gg);
      float h  = o_ * tanhf(cn);
      sC[r][j] = cn;
      sH[r][j] = h;
      int b = btile * 16 + r;
      Hout[((size_t)b * Tlen + pos) * (2 * Hdim) + dir * Hdim + j] = h;
    }
    __syncthreads();
  }
}

// ---------------------------------------------------------------------------
// Kernel 3: logits = [hf|hb](32768x100) @ fc_w^T(100x9) + fc_b, via WMMA,
// N padded 9 -> 16 (zero-row pointer). One wave per 16-row M tile.
// ---------------------------------------------------------------------------
__global__ void __launch_bounds__(32)
logits_kernel(const float* __restrict__ Hout, const float* __restrict__ fcw,
              const float* __restrict__ fcb, const float* __restrict__ zrow,
              float* __restrict__ out) {
  const int lane = threadIdx.x & 31;
  const int half = lane >> 4;
  const int ln   = lane & 15;
  const size_t mbase = (size_t)blockIdx.x * 16;
  const bool cv = ln < Cdim;
  const float* __restrict__ Crow = cv ? (fcw + (size_t)ln * (2 * Hdim)) : zrow;

  v8f acc = {};
  for (int k = 0; k < 2 * Hdim; k += 4) {            // K=100, 25 iters
    v2f a, b;
    a.x = Hout[(mbase + ln) * (2 * Hdim) + k + 2 * half];
    a.y = Hout[(mbase + ln) * (2 * Hdim) + k + 2 * half + 1];
    b.x = Crow[k + 2 * half];
    b.y = Crow[k + 2 * half + 1];
    acc = __builtin_amdgcn_wmma_f32_16x16x4_f32(false, a, false, b,
                                                (short)0, acc, false, false);
  }
  if (cv) {
    float bias = fcb[ln];
#pragma unroll
    for (int r = 0; r < 8; ++r)
      out[1 + (mbase + r + 8 * half) * Cdim + ln] = acc[r] + bias;
  }
}

// ---------------------------------------------------------------------------
// Kernel 4: CRF log-likelihood per batch row. One wave per b; lane j<9 owns
// alpha[j]; cross-lane gathers via __shfl (wave32).
// ---------------------------------------------------------------------------
__global__ void __launch_bounds__(32)
crf_kernel(const float* __restrict__ out, const int* __restrict__ ids,
           const int* __restrict__ labels, const float* __restrict__ trans,
           const float* __restrict__ startT, const float* __restrict__ endT,
           float* __restrict__ ll) {
  const int b    = blockIdx.x;
  const int lane = threadIdx.x;
  const float* __restrict__ logits = out + 1;
  const bool jv = lane < Cdim;
  const int  j  = jv ? lane : 0;
  const size_t base = (size_t)b * Tlen;

  float tr[Cdim];
#pragma unroll
  for (int i = 0; i < Cdim; ++i) tr[i] = trans[i * Cdim + j];

  float alpha = jv ? (startT[j] + logits[base * Cdim + j]) : -1e30f;

  // numerator (computed redundantly in all lanes; no divergence)
  int   prev = labels[base];
  float num  = startT[prev] + logits[base * Cdim + prev];
  int   last = prev;

  for (int t = 1; t < Tlen; ++t) {
    float et = jv ? logits[(base + t) * Cdim + j] : 0.f;
    float av[Cdim];
#pragma unroll
    for (int i = 0; i < Cdim; ++i) av[i] = __shfl(alpha, i, 32) + tr[i];
    float mx = av[0];
#pragma unroll
    for (int i = 1; i < Cdim; ++i) mx = fmaxf(mx, av[i]);
    float s = 0.f;
#pragma unroll
    for (int i = 0; i < Cdim; ++i) s += __expf(av[i] - mx);
    float nxt = mx + __logf(s) + et;

    const bool mt = ids[base + t] != 0;
    alpha = (mt && jv) ? nxt : alpha;

    int   tg  = labels[base + t];
    float mtf = mt ? 1.f : 0.f;
    num += mtf * (trans[prev * Cdim + tg] + logits[(base + t) * Cdim + tg]);
    if (mt) last = tg;
    prev = tg;
  }
  num += endT[last];

  float val = jv ? (alpha + endT[j]) : -1e30f;
  float m2 = -1e30f;
#pragma unroll
  for (int i = 0; i < Cdim; ++i) m2 = fmaxf(m2, __shfl(val, i, 32));
  float s2 = 0.f;
#pragma unroll
  for (int i = 0; i < Cdim; ++i) s2 += __expf(__shfl(val, i, 32) - m2);
  float logZ = m2 + __logf(s2);

  if (lane == 0) ll[b] = num - logZ;
}

// ---------------------------------------------------------------------------
// Kernel 5: loss = -mean(ll)  -> d_out[0]
// ---------------------------------------------------------------------------
__global__ void __launch_bounds__(32)
loss_kernel(const float* __restrict__ ll, float* __restrict__ out) {
  const int lane = threadIdx.x;
  float s = 0.f;
  for (int k = lane; k < Bsz; k += 32) s += ll[k];
#pragma unroll
  for (int off = 16; off > 0; off >>= 1) s += __shfl_down(s, off, 32);
  if (lane == 0) out[0] = -s / (float)Bsz;
}

// ---------------------------------------------------------------------------
extern "C" void kernel_launch(void* const* d_in, const int* in_sizes, int n_in,
                              void* d_out, int out_size, void* d_ws, size_t ws_size,
                              hipStream_t stream) {
  (void)in_sizes; (void)n_in; (void)out_size; (void)ws_size;

  const int*   ids    = (const int*)  d_in[0];
  const int*   labels = (const int*)  d_in[1];
  const float* emb    = (const float*)d_in[2];
  const float* WihF   = (const float*)d_in[3];
  const float* WhhF   = (const float*)d_in[4];
  const float* bihF   = (const float*)d_in[5];
  const float* bhhF   = (const float*)d_in[6];
  const float* WihB   = (const float*)d_in[7];
  const float* WhhB   = (const float*)d_in[8];
  const float* bihB   = (const float*)d_in[9];
  const float* bhhB   = (const float*)d_in[10];
  const float* fcw    = (const float*)d_in[11];
  const float* fcb    = (const float*)d_in[12];
  const float* trans  = (const float*)d_in[13];
  const float* startT = (const float*)d_in[14];
  const float* endT   = (const float*)d_in[15];
  float* out = (float*)d_out;

  float* XGF  = (float*)d_ws;                       // 32768*200
  float* XGB  = XGF + (size_t)Mrows * G4H;          // 32768*200
  float* Hout = XGB + (size_t)Mrows * G4H;          // 32768*100
  float* ll   = Hout + (size_t)Mrows * 2 * Hdim;    // 128
  float* zrow = ll + 128;                           // 512 zeros

  zrow_kernel<<<1, 512, 0, stream>>>(zrow);
  gates_kernel<<<Mrows / 16, 128, 0, stream>>>(ids, emb, WihF, bihF, bhhF,
                                               WihB, bihB, bhhB, zrow, XGF, XGB);
  lstm_kernel<<<dim3(Bsz / 16, 2), 128, 0, stream>>>(XGF, XGB, WhhF, WhhB, Hout);
  logits_kernel<<<Mrows / 16, 32, 0, stream>>>(Hout, fcw, fcb, zrow, out);
  crf_kernel<<<Bsz, 32, 0, stream>>>(out, ids, labels, trans, startT, endT, ll);
  loss_kernel<<<1, 32, 0, stream>>>(ll, out);
}